// RotationLayer_63041529970723
// MI455X (gfx1250) — compile-verified
//
#include <hip/hip_runtime.h>

#define DIM 4096
#define NPLANES 1024
#define NROWS (4 * 4096)             // B * S
#define ROWS_PER_BLOCK 16
#define NBLOCKS (NROWS / ROWS_PER_BLOCK)
#define TBL_BYTES (3 * DIM * 4)      // partner|ct|st = 48 KB
#define ROW_BYTES (DIM * 4)          // 16 KB
#define LDS_BYTES (TBL_BYTES + 2 * ROW_BYTES)  // 80 KB dynamic LDS

typedef float __attribute__((ext_vector_type(4))) f4;
typedef int   __attribute__((ext_vector_type(4))) i4;

// ---------------------------------------------------------------------------
// CDNA5 async-copy helpers (ASYNCcnt-tracked, no VGPR staging).
// ---------------------------------------------------------------------------
__device__ __forceinline__ void async_copy_b128(unsigned lds_addr, const void* gaddr) {
    asm volatile("global_load_async_to_lds_b128 %0, %1, off"
                 :: "v"(lds_addr), "v"((unsigned long long)(size_t)gaddr)
                 : "memory");
}
__device__ __forceinline__ void async_copy_b128_nt(unsigned lds_addr, const void* gaddr) {
    // Streaming read-once data: non-temporal so L2 keeps the coefficient tables.
    asm volatile("global_load_async_to_lds_b128 %0, %1, off th:TH_LOAD_NT"
                 :: "v"(lds_addr), "v"((unsigned long long)(size_t)gaddr)
                 : "memory");
}
__device__ __forceinline__ void wait_async0() {
    asm volatile("s_wait_asynccnt 0x0" ::: "memory");
}

// ---------------------------------------------------------------------------
// Table setup in d_ws: partner[DIM] (int), ct[DIM], st[DIM].
// out[d] = ct[d]*x[d] + st[d]*x[partner[d]] for every d.
// ---------------------------------------------------------------------------
__global__ __launch_bounds__(256) void init_table_kernel(int* __restrict__ partner,
                                                         float* __restrict__ ct,
                                                         float* __restrict__ st) {
    int d = blockIdx.x * 256 + threadIdx.x;
    if (d < DIM) {
        partner[d] = d;
        ct[d] = 1.0f;
        st[d] = 0.0f;
    }
}

__global__ __launch_bounds__(256) void fill_table_kernel(const float* __restrict__ angles,
                                                         const int* __restrict__ plane_i,
                                                         const int* __restrict__ plane_j,
                                                         int* __restrict__ partner,
                                                         float* __restrict__ ct,
                                                         float* __restrict__ st) {
    int k = blockIdx.x * 256 + threadIdx.x;
    if (k < NPLANES) {
        int i = plane_i[k];
        int j = plane_j[k];
        float a = angles[k];
        float c = __builtin_cosf(a);
        float s = __builtin_sinf(a);
        partner[i] = j; ct[i] = c; st[i] = -s;   // out[i] = c*x[i] - s*x[j]
        partner[j] = i; ct[j] = c; st[j] = s;    // out[j] = s*x[i] + c*x[j]
    }
}

// ---------------------------------------------------------------------------
// Main kernel: 16 rows per block. Tables async-copied to LDS once per block;
// rows double-buffered in LDS with ASYNCcnt software pipelining.
// LDS layout: [partner 16K][ct 16K][st 16K][rowbuf0 16K][rowbuf1 16K]
// ---------------------------------------------------------------------------
__global__ __launch_bounds__(256) void rot_rows_kernel(const float* __restrict__ x,
                                                       const void* __restrict__ tables,
                                                       float* __restrict__ out) {
    extern __shared__ unsigned char lds_all[];
    const int*   s_pn = (const int*)  (lds_all);
    const float* s_ct = (const float*)(lds_all + DIM * 4);
    const float* s_st = (const float*)(lds_all + 2 * DIM * 4);
    float*       bufs = (float*)      (lds_all + TBL_BYTES);

    const int t = threadIdx.x;
    const unsigned lds0 = (unsigned)(size_t)(void*)lds_all;  // LDS byte offset (low 32b of flat)
    const long long row0 = (long long)blockIdx.x * ROWS_PER_BLOCK;

    // Prologue: async-copy 48 KB of tables (12 x b128 per thread) ...
#pragma unroll
    for (int q = 0; q < 12; ++q) {
        const int byteoff = (q * 256 + t) * 16;
        async_copy_b128(lds0 + (unsigned)byteoff, (const unsigned char*)tables + byteoff);
    }
    // ... and row 0 into buffer 0 (4 x b128 per thread, NT).
#pragma unroll
    for (int q = 0; q < 4; ++q) {
        const int d = q * 1024 + t * 4;
        async_copy_b128_nt(lds0 + TBL_BYTES + (unsigned)(d * 4), x + row0 * DIM + d);
    }

    for (int r = 0; r < ROWS_PER_BLOCK; ++r) {
        const int cur = r & 1;
        const int nxt = cur ^ 1;

        wait_async0();     // this wave's outstanding copies (row r [+tables]) complete
        __syncthreads();   // all waves' copies done; buf[nxt]'s readers (row r-1) done

        // Kick off row r+1 into the other buffer; overlaps compute+store of row r.
        if (r + 1 < ROWS_PER_BLOCK) {
#pragma unroll
            for (int q = 0; q < 4; ++q) {
                const int d = q * 1024 + t * 4;
                async_copy_b128_nt(lds0 + TBL_BYTES + (unsigned)(nxt * ROW_BYTES + d * 4),
                                   x + (row0 + r + 1) * DIM + d);
            }
        }

        const float* __restrict__ rowbuf = bufs + cur * DIM;
        float* __restrict__ orow = out + (row0 + r) * DIM;
#pragma unroll
        for (int q = 0; q < 4; ++q) {
            const int d = q * 1024 + t * 4;
            f4 c = *(const f4*)(s_ct + d);          // ds_load_b128
            f4 s = *(const f4*)(s_st + d);          // ds_load_b128
            i4 p = *(const i4*)(s_pn + d);          // ds_load_b128
            f4 v = *(const f4*)(rowbuf + d);        // ds_load_b128
            f4 g;
            g.x = rowbuf[p.x];                      // ds_load_b32 gather x4
            g.y = rowbuf[p.y];
            g.z = rowbuf[p.z];
            g.w = rowbuf[p.w];
            f4 o = c * v + s * g;
            __builtin_nontemporal_store(o, (f4*)(orow + d));  // streaming global_store_b128
        }
    }
}

// ---------------------------------------------------------------------------
// Launch
// ---------------------------------------------------------------------------
extern "C" void kernel_launch(void* const* d_in, const int* in_sizes, int n_in,
                              void* d_out, int out_size, void* d_ws, size_t ws_size,
                              hipStream_t stream) {
    const float* x       = (const float*)d_in[0];
    const float* angles  = (const float*)d_in[1];
    const int*   plane_i = (const int*)d_in[2];
    const int*   plane_j = (const int*)d_in[3];
    float* out = (float*)d_out;

    int*   partner = (int*)d_ws;
    float* ct      = (float*)(partner + DIM);
    float* st      = ct + DIM;

    init_table_kernel<<<DIM / 256, 256, 0, stream>>>(partner, ct, st);
    fill_table_kernel<<<NPLANES / 256, 256, 0, stream>>>(angles, plane_i, plane_j,
                                                         partner, ct, st);
    rot_rows_kernel<<<NBLOCKS, 256, LDS_BYTES, stream>>>(x, (const void*)d_ws, out);
}